// GraphConvolution_16999480558222
// MI455X (gfx1250) — compile-verified
//
#include <hip/hip_runtime.h>

typedef __attribute__((ext_vector_type(2))) float v2f;
typedef __attribute__((ext_vector_type(8))) float v8f;

#define DF 256  // D_IN == D_OUT == 256

// ---------------------------------------------------------------------------
// Kernel 1: zero the aggregation workspace (must be re-zeroed every call).
// ---------------------------------------------------------------------------
__global__ __launch_bounds__(256) void gc_zero_kernel(float4* __restrict__ p, int n4) {
  int i = blockIdx.x * 256 + threadIdx.x;
  if (i < n4) p[i] = make_float4(0.f, 0.f, 0.f, 0.f);
}

// ---------------------------------------------------------------------------
// Kernel 2: scatter-aggregate. One block per face, thread = feature channel.
// Load each of the 3 vertex rows once, pre-sum neighbor pairs, then 3 f32
// atomics per channel (all L2-resident: aggr is 51.2MB < 192MB L2).
// ---------------------------------------------------------------------------
__global__ __launch_bounds__(256) void gc_scatter_kernel(
    const float* __restrict__ X, const long long* __restrict__ faces,
    float* __restrict__ aggr, int n_faces) {
  int f = blockIdx.x;
  if (f >= n_faces) return;
  int t = threadIdx.x;  // channel 0..255
  long long v1 = faces[(size_t)f * 3 + 0];
  long long v2 = faces[(size_t)f * 3 + 1];
  long long v3 = faces[(size_t)f * 3 + 2];
  float x1 = X[(size_t)v1 * DF + t];
  float x2 = X[(size_t)v2 * DF + t];
  float x3 = X[(size_t)v3 * DF + t];
  atomicAdd(&aggr[(size_t)v1 * DF + t], x2 + x3);
  atomicAdd(&aggr[(size_t)v2 * DF + t], x1 + x3);
  atomicAdd(&aggr[(size_t)v3 * DF + t], x1 + x2);
}

// ---------------------------------------------------------------------------
// Kernel 3: fused dual GEMM with V_WMMA_F32_16X16X4_F32.
//   out[m,n] = sum_k X[m,k]*W1[n,k] + sum_k A[m,k]*W2[n,k] + b1[n] + b2[n]
//
// One wave per 32(M) x 32(N) tile: two M sub-tiles share the B fragments,
// both GEMM paths accumulate into the same 4 accumulators (8 WMMAs / 8 b64
// loads per K-step). K-loop is software-pipelined: next step's fragments are
// fetched before this step's WMMAs so loads overlap the matrix pipe.
//
// f32 WMMA fragment layout (ISA 7.12.2):
//   A 16x4: lane l -> M = l%16 ; VGPR v -> K = v + 2*(l/16)  => float2 load
//   B 4x16: lane l -> N = l%16 ; VGPR v -> K = v + 2*(l/16)  => float2 load
//   C 16x16: VGPR r, lane l -> (M = r + 8*(l/16), N = l%16)
// ---------------------------------------------------------------------------
#define WMMA_F32(A, B, C) \
  __builtin_amdgcn_wmma_f32_16x16x4_f32(false, (A), false, (B), (short)0, (C), false, false)

__global__ __launch_bounds__(32) void gc_gemm_kernel(
    const float* __restrict__ X, const float* __restrict__ A,
    const float* __restrict__ W1, const float* __restrict__ W2,
    const float* __restrict__ b1, const float* __restrict__ b2,
    float* __restrict__ out, int n_verts) {
  const int lane = threadIdx.x;        // 0..31
  const int lid  = lane & 15;
  const int half = lane >> 4;          // 0 or 1 (selects K pair)
  const int m0 = blockIdx.x * 32;      // two 16-row sub-tiles: m0, m0+16
  const int n0 = blockIdx.y * 32;      // two 16-col sub-tiles: n0, n0+16

  // Clamp A-side rows so loads stay in-bounds without touching EXEC.
  int mA = m0 + lid;
  int mB = m0 + 16 + lid;
  int mcA = (mA < n_verts) ? mA : (n_verts - 1);
  int mcB = (mB < n_verts) ? mB : (n_verts - 1);
  const int n_a = n0 + lid;            // B columns for this lane (tile 0)
  const int n_b = n_a + 16;            // tile 1

  const int kofs = 2 * half;           // per-lane K offset within a step
  const float* x0 = X  + (size_t)mcA * DF + kofs;
  const float* x1 = X  + (size_t)mcB * DF + kofs;
  const float* a0 = A  + (size_t)mcA * DF + kofs;
  const float* a1 = A  + (size_t)mcB * DF + kofs;
  const float* w1a = W1 + (size_t)n_a * DF + kofs;
  const float* w1b = W1 + (size_t)n_b * DF + kofs;
  const float* w2a = W2 + (size_t)n_a * DF + kofs;
  const float* w2b = W2 + (size_t)n_b * DF + kofs;

  v8f c00 = {}, c01 = {}, c10 = {}, c11 = {};

  // Prologue: fragments for k = 0.
  v2f fx0 = *(const v2f*)(x0);
  v2f fx1 = *(const v2f*)(x1);
  v2f fa0 = *(const v2f*)(a0);
  v2f fa1 = *(const v2f*)(a1);
  v2f f1a = *(const v2f*)(w1a);
  v2f f1b = *(const v2f*)(w1b);
  v2f f2a = *(const v2f*)(w2a);
  v2f f2b = *(const v2f*)(w2b);

#pragma unroll 4
  for (int k = 4; k < DF; k += 4) {
    // Prefetch next step's fragments before consuming current ones.
    v2f nx0 = *(const v2f*)(x0 + k);
    v2f nx1 = *(const v2f*)(x1 + k);
    v2f na0 = *(const v2f*)(a0 + k);
    v2f na1 = *(const v2f*)(a1 + k);
    v2f n1a = *(const v2f*)(w1a + k);
    v2f n1b = *(const v2f*)(w1b + k);
    v2f n2a = *(const v2f*)(w2a + k);
    v2f n2b = *(const v2f*)(w2b + k);

    // 4 independent accumulator chains, depth 2 (X path then aggr path).
    c00 = WMMA_F32(fx0, f1a, c00);
    c01 = WMMA_F32(fx0, f1b, c01);
    c10 = WMMA_F32(fx1, f1a, c10);
    c11 = WMMA_F32(fx1, f1b, c11);
    c00 = WMMA_F32(fa0, f2a, c00);
    c01 = WMMA_F32(fa0, f2b, c01);
    c10 = WMMA_F32(fa1, f2a, c10);
    c11 = WMMA_F32(fa1, f2b, c11);

    fx0 = nx0; fx1 = nx1; fa0 = na0; fa1 = na1;
    f1a = n1a; f1b = n1b; f2a = n2a; f2b = n2b;
  }
  // Epilogue: last K-step.
  c00 = WMMA_F32(fx0, f1a, c00);
  c01 = WMMA_F32(fx0, f1b, c01);
  c10 = WMMA_F32(fx1, f1a, c10);
  c11 = WMMA_F32(fx1, f1b, c11);
  c00 = WMMA_F32(fa0, f2a, c00);
  c01 = WMMA_F32(fa0, f2b, c01);
  c10 = WMMA_F32(fa1, f2a, c10);
  c11 = WMMA_F32(fa1, f2b, c11);

  const float bias_a = b1[n_a] + b2[n_a];
  const float bias_b = b1[n_b] + b2[n_b];

#pragma unroll
  for (int r = 0; r < 8; ++r) {
    int row0 = m0 + r + 8 * half;        // C/D row for VGPR r, M sub-tile 0
    int row1 = row0 + 16;                // M sub-tile 1
    if (row0 < n_verts) {
      out[(size_t)row0 * DF + n_a] = c00[r] + bias_a;
      out[(size_t)row0 * DF + n_b] = c01[r] + bias_b;
    }
    if (row1 < n_verts) {
      out[(size_t)row1 * DF + n_a] = c10[r] + bias_a;
      out[(size_t)row1 * DF + n_b] = c11[r] + bias_b;
    }
  }
}

// ---------------------------------------------------------------------------
// Launcher. Inputs (setup_inputs order):
//   0: shape_features (N_VERTS*256 f32)   1: faces (N_FACES*3 int64)
//   2: W1 (256*256 f32)  3: b1 (256 f32)  4: W2 (256*256 f32)  5: b2 (256 f32)
// d_ws: first n_verts*256 floats used as the aggregation buffer.
// ---------------------------------------------------------------------------
extern "C" void kernel_launch(void* const* d_in, const int* in_sizes, int n_in,
                              void* d_out, int out_size, void* d_ws, size_t ws_size,
                              hipStream_t stream) {
  const float*     X     = (const float*)d_in[0];
  const long long* faces = (const long long*)d_in[1];
  const float*     W1    = (const float*)d_in[2];
  const float*     b1    = (const float*)d_in[3];
  const float*     W2    = (const float*)d_in[4];
  const float*     b2    = (const float*)d_in[5];
  float*           out   = (float*)d_out;
  float*           aggr  = (float*)d_ws;

  const int n_verts = in_sizes[0] / DF;
  const int n_faces = in_sizes[1] / 3;

  // 1) zero aggr
  int n4 = (n_verts * DF) / 4;
  gc_zero_kernel<<<(n4 + 255) / 256, 256, 0, stream>>>((float4*)aggr, n4);

  // 2) scatter-aggregate
  gc_scatter_kernel<<<n_faces, 256, 0, stream>>>(X, faces, aggr, n_faces);

  // 3) fused dual GEMM + bias (32x32 tile per wave)
  dim3 grid((n_verts + 31) / 32, DF / 32);
  gc_gemm_kernel<<<grid, 32, 0, stream>>>(X, aggr, W1, W2, b1, b2, out, n_verts);
}